// RWKV_TimeMix_420906795382
// MI455X (gfx1250) — compile-verified
//
#include <hip/hip_runtime.h>
#include <hip/hip_bf16.h>

typedef float v2f __attribute__((ext_vector_type(2)));
typedef float v4f __attribute__((ext_vector_type(4)));
typedef float v8f __attribute__((ext_vector_type(8)));

namespace {
constexpr int B_  = 8;
constexpr int T_  = 2048;
constexpr int C_  = 1024;
constexpr int D_  = 1024;
constexpr int BT_ = B_ * T_;

// GEMM tiling: block tile 64(M) x 64(N), K-stage 32; 8 waves (wave32), each 16x32.
constexpr int MT  = 64;
constexpr int NT  = 64;
constexpr int KB  = 32;
constexpr int LDT = KB + 4;   // LDS row pitch (floats): float4 stores stay 16B aligned

// chunked WKV scan
constexpr int NCH = 16;
constexpr int TC  = T_ / NCH; // 128
}

__device__ __forceinline__ float sigmoidf_(float x) {
  return 1.0f / (1.0f + __expf(-x));
}

// ---------------------------------------------------------------------------
// CDNA5 async global->LDS copy (ASYNCcnt path), 16 bytes per lane.
// VDST = 32-bit LDS byte offset, VADDR = 64-bit global address, SADDR = off.
// Inline asm per CDNA5 ISA (portable across toolchains).
// ---------------------------------------------------------------------------
__device__ __forceinline__ void async_copy16(const float* lds_dst, const float* g_src) {
  unsigned int      loff = (unsigned int)(size_t)lds_dst;  // low 32 bits = LDS offset
  unsigned long long ga  = (unsigned long long)(size_t)g_src;
  asm volatile("global_load_async_to_lds_b128 %0, %1, off"
               :: "v"(loff), "v"(ga) : "memory");
}

__device__ __forceinline__ void async_wait_all() {
  asm volatile("s_wait_asynccnt 0x0" ::: "memory");
}

// ---------------------------------------------------------------------------
// Double-buffered WMMA f32 GEMM core:  out[m, n] = sum_k A'[m,k] * W[n,k]
//   MIX:  A'[m,k] = A[m,k]*tm[k] + A[m-1,k]*(1-tm[k])   (time-shift, reg path)
//   !MIX: A' = A, staged with async global->LDS copies
// B (weights) always staged with async global->LDS copies.
// ---------------------------------------------------------------------------
template<bool MIX>
__device__ __forceinline__ void gemm_core(
    const float* __restrict__ Ag,   // [rows, K]
    const float* __restrict__ Wg,   // [Nout, K]
    const float* __restrict__ tm,   // mixing vector [K] (MIX only)
    float* __restrict__ out,        // [rows, Nout]
    int m0, int n0, int K, int Nout, bool dosig)
{
  __shared__ float As[2][MT][LDT];
  __shared__ float Bs[2][NT][LDT];

  const int tid  = threadIdx.x;
  const int lane = tid & 31;
  const int wave = tid >> 5;               // 0..7
  const int wm   = wave & 3;               // 4 waves along M
  const int wn   = wave >> 2;              // 2 waves along N
  const int lrow = tid >> 3;               // 0..31 tile row per load pass
  const int lcol = (tid & 7) * 4;          // 0,4,...,28

  v8f acc0 = {}, acc1 = {};
  v4f rA[2];

  auto issueB = [&](int k0, int buf) {
    #pragma unroll
    for (int p = 0; p < 2; ++p) {
      const int n = p * 32 + lrow;
      async_copy16(&Bs[buf][n][lcol], Wg + (size_t)(n0 + n) * K + k0 + lcol);
    }
  };
  auto issueA = [&](int k0, int buf) {   // !MIX: pure async copy
    #pragma unroll
    for (int p = 0; p < 2; ++p) {
      const int m = p * 32 + lrow;
      async_copy16(&As[buf][m][lcol], Ag + (size_t)(m0 + m) * K + k0 + lcol);
    }
  };
  auto loadA_mix = [&](int k0) {          // MIX: time-shift mixing in registers
    #pragma unroll
    for (int p = 0; p < 2; ++p) {
      const int m   = p * 32 + lrow;
      const int row = m0 + m;
      const int t   = row & (T_ - 1);
      const float* xr = Ag + (size_t)row * K + k0 + lcol;
      v4f cur = *(const v4f*)xr;
      v4f prv = {};
      if (t > 0) prv = *(const v4f*)(xr - K);
      v4f mu = *(const v4f*)(tm + k0 + lcol);
      v4f mx;
      mx.x = cur.x * mu.x + prv.x * (1.0f - mu.x);
      mx.y = cur.y * mu.y + prv.y * (1.0f - mu.y);
      mx.z = cur.z * mu.z + prv.z * (1.0f - mu.z);
      mx.w = cur.w * mu.w + prv.w * (1.0f - mu.w);
      rA[p] = mx;
    }
  };
  auto storeA = [&](int buf) {
    #pragma unroll
    for (int p = 0; p < 2; ++p)
      *(v4f*)&As[buf][p * 32 + lrow][lcol] = rA[p];
  };

  // ---- prologue: fill buffer 0 ----
  if (MIX) { loadA_mix(0); storeA(0); } else { issueA(0, 0); }
  issueB(0, 0);
  async_wait_all();
  __syncthreads();

  int buf = 0;
  for (int k0 = 0; k0 < K; k0 += KB) {
    const bool haveNext = (k0 + KB) < K;
    if (haveNext) {                       // kick next stage into buf^1
      if (MIX) loadA_mix(k0 + KB); else issueA(k0 + KB, buf ^ 1);
      issueB(k0 + KB, buf ^ 1);
    }

    // ---- WMMA mainloop on current buffer (ISA 7.12.2 fragment layouts) ----
    const int am   = wm * 16 + (lane & 15);
    const int bn   = wn * 32 + (lane & 15);
    const int koff = (lane >> 4) * 2;     // lanes 0-15 -> K pair 0/1, 16-31 -> 2/3
    #pragma unroll
    for (int kk = 0; kk < KB; kk += 4) {
      v2f a  = *(const v2f*)&As[buf][am][kk + koff];
      v2f b0 = *(const v2f*)&Bs[buf][bn][kk + koff];
      v2f b1 = *(const v2f*)&Bs[buf][bn + 16][kk + koff];
      acc0 = __builtin_amdgcn_wmma_f32_16x16x4_f32(false, a, false, b0,
                                                   (short)0, acc0, false, false);
      acc1 = __builtin_amdgcn_wmma_f32_16x16x4_f32(false, a, false, b1,
                                                   (short)0, acc1, false, false);
    }

    if (haveNext && MIX) storeA(buf ^ 1); // reg->LDS after compute kicked off
    async_wait_all();
    __syncthreads();
    buf ^= 1;
  }

  // ---- epilogue: C/D layout (VGPR r -> M=r or r+8, lane[4] splits M halves) ----
  const int mrow = m0 + wm * 16 + ((lane >> 4) << 3);
  const int ncol = n0 + wn * 32 + (lane & 15);
  #pragma unroll
  for (int r = 0; r < 8; ++r) {
    float y0 = acc0[r];
    float y1 = acc1[r];
    if (dosig) { y0 = sigmoidf_(y0); y1 = sigmoidf_(y1); }
    const size_t base = (size_t)(mrow + r) * Nout;
    out[base + ncol]      = y0;
    out[base + ncol + 16] = y1;
  }
}

// ---------------------------------------------------------------------------
// Kernel 1: fused k/v/sigmoid(r) projections. grid = (D/NT, BT/MT, 3)
// ---------------------------------------------------------------------------
__global__ __launch_bounds__(256)
void kvr_gemm(const float* __restrict__ x,
              const float* __restrict__ tmk, const float* __restrict__ tmv,
              const float* __restrict__ tmr,
              const float* __restrict__ Wk, const float* __restrict__ Wv,
              const float* __restrict__ Wr,
              float* __restrict__ ok, float* __restrict__ ov,
              float* __restrict__ osr)
{
  const int mi = blockIdx.z;
  const float* W   = (mi == 0) ? Wk  : (mi == 1) ? Wv  : Wr;
  const float* tm  = (mi == 0) ? tmk : (mi == 1) ? tmv : tmr;
  float*       out = (mi == 0) ? ok  : (mi == 1) ? ov  : osr;
  gemm_core<true>(x, W, tm, out, blockIdx.y * MT, blockIdx.x * NT, C_, D_, mi == 2);
}

// ---------------------------------------------------------------------------
// Kernel 3: output projection  out[row, c] = sum_d rwkv[row, d] * Wo[c, d]
// ---------------------------------------------------------------------------
__global__ __launch_bounds__(256)
void out_gemm(const float* __restrict__ A, const float* __restrict__ Wo,
              float* __restrict__ out)
{
  gemm_core<false>(A, Wo, nullptr, out, blockIdx.y * MT, blockIdx.x * NT, D_, C_, false);
}

// ---------------------------------------------------------------------------
// Kernel 2a: per-chunk WKV carries. One thread per (b, chunk, d).
// ---------------------------------------------------------------------------
__global__ __launch_bounds__(256)
void wkv_pass1(const float* __restrict__ k, const float* __restrict__ v,
               const float* __restrict__ time_decay,
               float* __restrict__ carryA, float* __restrict__ carryB)
{
  const int gid = blockIdx.x * 256 + threadIdx.x;   // (b*NCH + ch)*D + d
  const int d  = gid & (D_ - 1);
  const int ch = (gid >> 10) & (NCH - 1);
  const int b  = gid >> 14;
  const float ew = __expf(-__expf(time_decay[d]));
  float a = 0.0f, bb = 0.0f;
  size_t idx = ((size_t)b * T_ + (size_t)ch * TC) * D_ + d;
  for (int t0 = 0; t0 < TC; t0 += 16) {
    __builtin_prefetch(&k[idx + 16 * (size_t)D_], 0, 1);
    __builtin_prefetch(&v[idx + 16 * (size_t)D_], 0, 1);
    #pragma unroll
    for (int t = 0; t < 16; ++t, idx += D_) {
      const float ek  = __expf(k[idx]);
      const float ekv = ek * v[idx];
      a  = ew * a  + ekv;
      bb = ew * bb + ek;
    }
  }
  carryA[gid] = a;
  carryB[gid] = bb;
}

// ---------------------------------------------------------------------------
// Kernel 2b: sequential prefix over the 16 chunk carries -> chunk init states.
// ---------------------------------------------------------------------------
__global__ __launch_bounds__(256)
void wkv_prefix(const float* __restrict__ time_decay,
                float* __restrict__ carryA, float* __restrict__ carryB)
{
  const int gid = blockIdx.x * 256 + threadIdx.x;   // b*D + d
  const int d = gid & (D_ - 1);
  const int b = gid >> 10;
  const float w   = -__expf(time_decay[d]);
  const float ewT = __expf(w * (float)TC);          // ew^TC in closed form
  float a = 0.0f, bb = 0.0f;
  #pragma unroll
  for (int ch = 0; ch < NCH; ++ch) {
    const size_t idx = ((size_t)b * NCH + ch) * D_ + d;
    const float ca = carryA[idx];
    const float cb = carryB[idx];
    carryA[idx] = a;       // replace carry with init state for this chunk
    carryB[idx] = bb;
    a  = ewT * a  + ca;
    bb = ewT * bb + cb;
  }
}

// ---------------------------------------------------------------------------
// Kernel 2c: replay each chunk from its init state, emit rwkv = sr * wkv.
// ---------------------------------------------------------------------------
__global__ __launch_bounds__(256)
void wkv_pass2(const float* __restrict__ k, const float* __restrict__ v,
               const float* __restrict__ sr,
               const float* __restrict__ time_decay,
               const float* __restrict__ time_first,
               const float* __restrict__ initA, const float* __restrict__ initB,
               float* __restrict__ rwkv)
{
  const int gid = blockIdx.x * 256 + threadIdx.x;   // (b*NCH + ch)*D + d
  const int d  = gid & (D_ - 1);
  const int ch = (gid >> 10) & (NCH - 1);
  const int b  = gid >> 14;
  const float ew = __expf(-__expf(time_decay[d]));
  const float eu = __expf(time_first[d]);
  float a  = initA[gid];
  float bb = initB[gid];
  size_t idx = ((size_t)b * T_ + (size_t)ch * TC) * D_ + d;
  for (int t0 = 0; t0 < TC; t0 += 16) {
    __builtin_prefetch(&k[idx + 16 * (size_t)D_], 0, 1);
    __builtin_prefetch(&v[idx + 16 * (size_t)D_], 0, 1);
    __builtin_prefetch(&sr[idx + 16 * (size_t)D_], 0, 1);
    #pragma unroll
    for (int t = 0; t < 16; ++t, idx += D_) {
      const float ek  = __expf(k[idx]);
      const float ekv = ek * v[idx];
      const float o   = (eu * ekv + a) / (eu * ek + bb);
      rwkv[idx] = sr[idx] * o;
      a  = ew * a  + ekv;
      bb = ew * bb + ek;
    }
  }
}

// ---------------------------------------------------------------------------
extern "C" void kernel_launch(void* const* d_in, const int* in_sizes, int n_in,
                              void* d_out, int out_size, void* d_ws, size_t ws_size,
                              hipStream_t stream) {
  const float* x          = (const float*)d_in[0];
  const float* time_decay = (const float*)d_in[1];
  const float* time_first = (const float*)d_in[2];
  const float* tmk        = (const float*)d_in[3];
  const float* tmv        = (const float*)d_in[4];
  const float* tmr        = (const float*)d_in[5];
  const float* Wk         = (const float*)d_in[6];
  const float* Wv         = (const float*)d_in[7];
  const float* Wr         = (const float*)d_in[8];
  const float* Wo         = (const float*)d_in[9];
  float* out = (float*)d_out;

  float* ws = (float*)d_ws;
  const size_t n = (size_t)BT_ * D_;          // 16M floats per buffer
  float* wk  = ws;
  float* wv  = ws + n;
  float* wsr = ws + 2 * n;
  float* wrk = ws + 3 * n;                    // rwkv
  float* cA  = ws + 4 * n;                    // B*NCH*D carries / inits
  float* cB  = cA + (size_t)B_ * NCH * D_;

  // 1) fused k/v/sigmoid(r) projection GEMMs
  dim3 g1(D_ / NT, BT_ / MT, 3);
  kvr_gemm<<<g1, 256, 0, stream>>>(x, tmk, tmv, tmr, Wk, Wv, Wr, wk, wv, wsr);

  // 2) chunked WKV scan
  const int scanThreads = B_ * NCH * D_;      // 131072
  wkv_pass1<<<scanThreads / 256, 256, 0, stream>>>(wk, wv, time_decay, cA, cB);
  wkv_prefix<<<(B_ * D_) / 256, 256, 0, stream>>>(time_decay, cA, cB);
  wkv_pass2<<<scanThreads / 256, 256, 0, stream>>>(wk, wv, wsr, time_decay,
                                                   time_first, cA, cB, wrk);

  // 3) output projection
  dim3 g3(C_ / NT, BT_ / MT, 1);
  out_gemm<<<g3, 256, 0, stream>>>(wrk, Wo, out);
}